// InterFormerEncoder_85495618995110
// MI455X (gfx1250) — compile-verified
//
#include <hip/hip_runtime.h>
#include <hip/hip_bf16.h>

#define BB 4
#define CC 200
#define WW 300
#define HH 128
#define NHH 8
#define PHH 16
#define FFF 512
#define LL 2
#define PEN 512
#define MAXLEN_ 200

typedef float v2f __attribute__((ext_vector_type(2)));
typedef float v8f __attribute__((ext_vector_type(8)));

// ---------------------------------------------------------------------------
// Sinusoidal positional encoding table: PE[512][128]
// ---------------------------------------------------------------------------
__global__ __launch_bounds__(256) void if_pe_fill(float* __restrict__ PE) {
    int t = blockIdx.x * blockDim.x + threadIdx.x;   // 512*64 threads
    if (t >= PEN * 64) return;
    int p = t >> 6;
    int j = t & 63;
    float freq = __expf(-(float)j * (logf(10000.0f) / 63.0f));
    float ang  = (float)p * freq;
    PE[p * HH + j]      = __sinf(ang);
    PE[p * HH + 64 + j] = __cosf(ang);
}

// ---------------------------------------------------------------------------
// LDS-tiled fp32 WMMA GEMM:  D[M,N] = (relu?) (A[M,K] @ B[K,N] + bias[N])
// 256 threads = 8 waves; macro-tile 64(M) x 32(N); K staged in slabs of 16.
// Each wave owns one 16x16 subtile (waves arranged 4x2) and accumulates with
// v_wmma_f32_16x16x4_f32, reading fragments from LDS (ds_load_b64 pairs).
// Requires N%32==0, K%16==0; M arbitrary (clamped loads, predicated stores).
// ---------------------------------------------------------------------------
__global__ __launch_bounds__(256) void if_gemm_wmma(
    const float* __restrict__ A, const float* __restrict__ Bw,
    const float* __restrict__ bias, float* __restrict__ D,
    int M, int N, int K, int relu)
{
    __shared__ __attribute__((aligned(16))) float As[64][20]; // [m][k], stride 20
    __shared__ __attribute__((aligned(16))) float Bs[32][20]; // [n][k], stride 20

    int tid  = threadIdx.x;
    int wid  = tid >> 5, lane = tid & 31;
    int half = lane >> 4, l = lane & 15;
    int wm = wid & 3, wn = wid >> 2;          // wave grid 4 x 2
    int m0 = blockIdx.y * 64;
    int n0 = blockIdx.x * 32;

    int mfrag = wm * 16 + l;                  // A row within tile (this lane)
    int nfrag = wn * 16 + l;                  // B col within tile (this lane)
    int gcol  = n0 + nfrag;

    float bcol = bias ? bias[gcol] : 0.0f;
    v8f c;
#pragma unroll
    for (int j = 0; j < 8; ++j) c[j] = bcol;

    // A loader: thread t -> row t/4 (0..63), k chunk (t%4)*4
    int la_row = tid >> 2;
    int la_k   = (tid & 3) * 4;
    int arow   = min(m0 + la_row, M - 1);     // clamp for partial tiles

    for (int k0 = 0; k0 < K; k0 += 16) {
        float4 av = *(const float4*)(A + (size_t)arow * K + k0 + la_k);
        *(float4*)&As[la_row][la_k] = av;
        if (tid < 128) {                      // B tile 16x32, transpose into LDS
            int bk = tid >> 3;                // 0..15
            int bn = (tid & 7) * 4;           // 0,4,...,28
            float4 bv = *(const float4*)(Bw + (size_t)(k0 + bk) * N + n0 + bn);
            Bs[bn + 0][bk] = bv.x;
            Bs[bn + 1][bk] = bv.y;
            Bs[bn + 2][bk] = bv.z;
            Bs[bn + 3][bk] = bv.w;
        }
        __syncthreads();
#pragma unroll
        for (int kk = 0; kk < 16; kk += 4) {
            // fp32 16x16x4 operand layout: lanes 0-15 -> K=kk,kk+1 ;
            // lanes 16-31 -> K=kk+2,kk+3 (both A and B fragments)
            v2f a = *(const v2f*)&As[mfrag][kk + 2 * half];
            v2f b = *(const v2f*)&Bs[nfrag][kk + 2 * half];
            c = __builtin_amdgcn_wmma_f32_16x16x4_f32(false, a, false, b,
                                                      (short)0, c, false, false);
        }
        __syncthreads();
    }

    int rbase = m0 + wm * 16 + 8 * half;
#pragma unroll
    for (int j = 0; j < 8; ++j) {
        int r = rbase + j;
        if (r < M) {
            float v = c[j];
            if (relu) v = fmaxf(v, 0.0f);
            D[(size_t)r * N + gcol] = v;
        }
    }
}

// ---------------------------------------------------------------------------
// Tqr[b,c,n,h] = sum_p (q[b,c,n*16+p] + vb[n*16+p]) * Wr[h, n*16+p]
// dterm[b,c,n] = sum_p (q[b,c,n*16+p] + vb[n*16+p]) * br[n*16+p]
// ---------------------------------------------------------------------------
__global__ __launch_bounds__(256) void if_tqr(
    const float* __restrict__ q, const float* __restrict__ vb_flat,
    const float* __restrict__ Wr, const float* __restrict__ br,
    float* __restrict__ Tqr, float* __restrict__ dterm)
{
    int t = blockIdx.x * blockDim.x + threadIdx.x;   // B*C*NH*H
    if (t >= BB * CC * NHH * HH) return;
    int h  = t & (HH - 1);
    int n  = (t >> 7) & (NHH - 1);
    int bc = t >> 10;

    const float* qrow = q + (size_t)bc * HH + n * PHH;
    const float* wrow = Wr + (size_t)h * HH + n * PHH;
    const float* vb   = vb_flat + n * PHH;

    float s = 0.0f;
#pragma unroll
    for (int p = 0; p < PHH; ++p) s += (qrow[p] + vb[p]) * wrow[p];
    Tqr[t] = s;

    if (h == 0) {
        const float* brp = br + n * PHH;
        float d = 0.0f;
#pragma unroll
        for (int p = 0; p < PHH; ++p) d += (qrow[p] + vb[p]) * brp[p];
        dterm[bc * NHH + n] = d;
    }
}

// ---------------------------------------------------------------------------
// Fused attention: per (b,c) block.  Tqr tile staged to LDS with the gfx1250
// async global->LDS engine (ASYNCcnt).
// ---------------------------------------------------------------------------
__global__ __launch_bounds__(256) void if_attn(
    const float* __restrict__ q,  const float* __restrict__ k,
    const float* __restrict__ v,  const float* __restrict__ u_flat,
    const float* __restrict__ Tqr, const float* __restrict__ dterm,
    const float* __restrict__ Ttab, const float* __restrict__ bf,
    const int* __restrict__ pos_s, const int* __restrict__ pos_e,
    const int* __restrict__ lex_s, const int* __restrict__ lex_e,
    const int* __restrict__ seq_len, const int* __restrict__ lex_num,
    float* __restrict__ ctx)
{
    __shared__ __attribute__((aligned(16))) float qu_s[HH];
    __shared__ __attribute__((aligned(16))) float tq_s[NHH][HH];
    __shared__ __attribute__((aligned(16))) float sc_s[NHH][WW + 4];
    __shared__ float dt_s[NHH];

    int bc = blockIdx.x;
    int b = bc / CC, c = bc % CC;
    int tid  = threadIdx.x;
    int wid  = tid >> 5;
    int lane = tid & 31;

    // Async copy: Tqr[bc] (1024 floats) -> tq_s.  One B128 per thread.
    {
        unsigned ldsoff = (unsigned)(size_t)(&tq_s[0][0]) + (unsigned)(tid * 16);
        const float* gsrc = Tqr + (size_t)bc * (NHH * HH) + tid * 4;
        asm volatile("global_load_async_to_lds_b128 %0, %1, off"
                     :: "v"(ldsoff), "v"(gsrc) : "memory");
    }
    if (tid < HH) qu_s[tid] = q[(size_t)bc * HH + tid] + u_flat[tid];
    if (tid < NHH) dt_s[tid] = dterm[bc * NHH + tid];
    asm volatile("s_wait_asynccnt 0x0" ::: "memory");
    __syncthreads();

    int  WL     = lex_num[b];
    bool cvalid = (c < seq_len[b]);
    int  ps = pos_s[b * CC + c] + MAXLEN_;
    int  pe = pos_e[b * CC + c] + MAXLEN_;

    if (cvalid) {
        int h0 = lane * 4;
        int myhead = lane >> 2;     // 4 lanes cover one head's 16 dims
        const float4 bf4 = *(const float4*)(bf + h0);
        float4 qu4 = *(const float4*)(&qu_s[h0]);
        float4 tq4[NHH];
#pragma unroll
        for (int n = 0; n < NHH; ++n) tq4[n] = *(const float4*)(&tq_s[n][h0]);

        for (int w = wid; w < WL; w += 8) {
            int ls = lex_s[b * WW + w], le = lex_e[b * WW + w];
            int i1 = min(max(ps - ls, 0), PEN - 1);
            int i2 = min(max(ps - le, 0), PEN - 1);
            int i3 = min(max(pe - ls, 0), PEN - 1);
            int i4 = min(max(pe - le, 0), PEN - 1);
            float4 t1 = *(const float4*)(Ttab + (size_t)i1 * HH + h0);
            float4 t2 = *(const float4*)(Ttab + (size_t)(PEN * HH)     + (size_t)i2 * HH + h0);
            float4 t3 = *(const float4*)(Ttab + (size_t)(2 * PEN * HH) + (size_t)i3 * HH + h0);
            float4 t4 = *(const float4*)(Ttab + (size_t)(3 * PEN * HH) + (size_t)i4 * HH + h0);
            float r0 = fmaxf(t1.x + t2.x + t3.x + t4.x + bf4.x, 0.0f);
            float r1 = fmaxf(t1.y + t2.y + t3.y + t4.y + bf4.y, 0.0f);
            float r2 = fmaxf(t1.z + t2.z + t3.z + t4.z + bf4.z, 0.0f);
            float r3 = fmaxf(t1.w + t2.w + t3.w + t4.w + bf4.w, 0.0f);

            float4 kk = *(const float4*)(k + ((size_t)b * WW + w) * HH + h0);
            float ac = qu4.x * kk.x + qu4.y * kk.y + qu4.z * kk.z + qu4.w * kk.w;

            float acc[NHH];
#pragma unroll
            for (int n = 0; n < NHH; ++n) {
                acc[n] = r0 * tq4[n].x + r1 * tq4[n].y + r2 * tq4[n].z + r3 * tq4[n].w;
                acc[n] += (myhead == n) ? ac : 0.0f;
            }
#pragma unroll
            for (int off = 16; off >= 1; off >>= 1)
#pragma unroll
                for (int n = 0; n < NHH; ++n)
                    acc[n] += __shfl_xor(acc[n], off, 32);
            if (lane < NHH) sc_s[lane][w] = acc[lane] + dt_s[lane];
        }
    }
    __syncthreads();

    // wave n handles head n: softmax over w<WL, then context
    int n = wid;
    float* out_ctx = ctx + (size_t)bc * HH + n * PHH;
    if (!cvalid) {
        if (lane < PHH) out_ctx[lane] = 0.0f;
        return;
    }
    float mx = -3.0e38f;
    for (int w = lane; w < WL; w += 32) mx = fmaxf(mx, sc_s[n][w]);
#pragma unroll
    for (int off = 16; off >= 1; off >>= 1) mx = fmaxf(mx, __shfl_xor(mx, off, 32));
    float sum = 0.0f;
    for (int w = lane; w < WL; w += 32) {
        float e = __expf(sc_s[n][w] - mx);
        sc_s[n][w] = e;
        sum += e;
    }
#pragma unroll
    for (int off = 16; off >= 1; off >>= 1) sum += __shfl_xor(sum, off, 32);
    float inv = 1.0f / sum;

    int p = lane & 15, w0 = lane >> 4;   // lane -> (p, w parity)
    float a = 0.0f;
    for (int w = w0; w < WL; w += 2)
        a += sc_s[n][w] * v[((size_t)b * WW + w) * HH + n * PHH + p];
    a += __shfl_xor(a, 16, 32);
    if (lane < PHH) out_ctx[p] = a * inv;
}

// ---------------------------------------------------------------------------
// out[row] = LayerNorm(x[row] + res[row]) * g + b    (one 128-wide row/block)
// ---------------------------------------------------------------------------
__global__ __launch_bounds__(128) void if_add_ln(
    const float* __restrict__ x, const float* __restrict__ res,
    const float* __restrict__ g, const float* __restrict__ bb,
    float* __restrict__ out)
{
    __shared__ float s1[4], s2[4];
    int row = blockIdx.x, tid = threadIdx.x;
    int wid = tid >> 5, lane = tid & 31;

    float v = x[(size_t)row * HH + tid] + res[(size_t)row * HH + tid];
    float s = v;
#pragma unroll
    for (int off = 16; off >= 1; off >>= 1) s += __shfl_xor(s, off, 32);
    if (lane == 0) s1[wid] = s;
    __syncthreads();
    float mean = (s1[0] + s1[1] + s1[2] + s1[3]) * (1.0f / HH);
    float d = v - mean;
    float q = d * d;
#pragma unroll
    for (int off = 16; off >= 1; off >>= 1) q += __shfl_xor(q, off, 32);
    if (lane == 0) s2[wid] = q;
    __syncthreads();
    float var = (s2[0] + s2[1] + s2[2] + s2[3]) * (1.0f / HH);
    out[(size_t)row * HH + tid] = d * rsqrtf(var + 1e-5f) * g[tid] + bb[tid];
}

// ---------------------------------------------------------------------------
extern "C" void kernel_launch(void* const* d_in, const int* in_sizes, int n_in,
                              void* d_out, int out_size, void* d_ws, size_t ws_size,
                              hipStream_t stream) {
    const float* chars  = (const float*)d_in[0];
    const float* words  = (const float*)d_in[1];
    const int* pos_s    = (const int*)d_in[2];
    const int* pos_e    = (const int*)d_in[3];
    const int* lex_s    = (const int*)d_in[4];
    const int* lex_e    = (const int*)d_in[5];
    const int* seq_len  = (const int*)d_in[6];
    const int* lex_num  = (const int*)d_in[7];
    const float* Wf     = (const float*)d_in[8];
    const float* bf     = (const float*)d_in[9];
    const float* Wq     = (const float*)d_in[10];
    const float* bq     = (const float*)d_in[11];
    const float* Wk     = (const float*)d_in[12];
    const float* bk     = (const float*)d_in[13];
    const float* Wv     = (const float*)d_in[14];
    const float* bv     = (const float*)d_in[15];
    const float* Wr     = (const float*)d_in[16];
    const float* br     = (const float*)d_in[17];
    const float* u_bias = (const float*)d_in[18];
    const float* v_bias = (const float*)d_in[19];
    const float* ln1_g  = (const float*)d_in[20];
    const float* ln1_b  = (const float*)d_in[21];
    const float* ln2_g  = (const float*)d_in[22];
    const float* ln2_b  = (const float*)d_in[23];
    const float* W1     = (const float*)d_in[24];
    const float* b1     = (const float*)d_in[25];
    const float* W2     = (const float*)d_in[26];
    const float* b2     = (const float*)d_in[27];
    float* out = (float*)d_out;
    float* ws  = (float*)d_ws;

    // workspace layout (floats)
    float* PEb  = ws;                         // 512*128      = 65536
    float* Tb   = PEb  + PEN * HH;            // 4*512*128    = 262144
    float* qb   = Tb   + 4 * PEN * HH;        // B*C*H        = 102400
    float* kb   = qb   + BB * CC * HH;        // B*W*H        = 153600
    float* vb   = kb   + BB * WW * HH;        // B*W*H        = 153600
    float* tqrb = vb   + BB * WW * HH;        // B*C*NH*H     = 819200
    float* dtb  = tqrb + BB * CC * NHH * HH;  // B*C*NH       = 6400
    float* ctxb = dtb  + BB * CC * NHH;       // B*C*H        = 102400
    float* cur  = ctxb + BB * CC * HH;        // B*C*H        = 102400
    float* f1b  = cur  + BB * CC * HH;        // B*C*FF       = 409600
    float* f2b  = f1b  + BB * CC * FFF;       // B*C*H        = 102400

    // 0) PE table + the four PE @ Wf[m] tables (index-independent)
    if_pe_fill<<<(PEN * 64 + 255) / 256, 256, 0, stream>>>(PEb);
    for (int m = 0; m < 4; ++m) {
        if_gemm_wmma<<<dim3(HH / 32, (PEN + 63) / 64), 256, 0, stream>>>(
            PEb, Wf + (size_t)m * HH * HH, nullptr, Tb + (size_t)m * PEN * HH,
            PEN, HH, HH, 0);
    }

    // out = chars
    hipMemcpyAsync(cur, chars, (size_t)BB * CC * HH * sizeof(float),
                   hipMemcpyDeviceToDevice, stream);

    for (int i = 0; i < LL; ++i) {
        const size_t wHH = (size_t)i * HH * HH;
        // q / k / v projections (WMMA GEMMs)
        if_gemm_wmma<<<dim3(HH / 32, (BB * CC + 63) / 64), 256, 0, stream>>>(
            cur, Wq + wHH, bq + i * HH, qb, BB * CC, HH, HH, 0);
        if_gemm_wmma<<<dim3(HH / 32, (BB * WW + 63) / 64), 256, 0, stream>>>(
            words, Wk + wHH, bk + i * HH, kb, BB * WW, HH, HH, 0);
        if_gemm_wmma<<<dim3(HH / 32, (BB * WW + 63) / 64), 256, 0, stream>>>(
            words, Wv + wHH, bv + i * HH, vb, BB * WW, HH, HH, 0);

        // factored Wr projection of (q + v_bias)
        if_tqr<<<(BB * CC * NHH * HH + 255) / 256, 256, 0, stream>>>(
            qb, v_bias + i * HH, Wr + wHH, br + i * HH, tqrb, dtb);

        // fused rel-gather + attention + softmax + context
        if_attn<<<BB * CC, 256, 0, stream>>>(
            qb, kb, vb, u_bias + i * HH, tqrb, dtb, Tb, bf,
            pos_s, pos_e, lex_s, lex_e, seq_len, lex_num, ctxb);

        // out = LN1(ctx + out)
        if_add_ln<<<BB * CC, 128, 0, stream>>>(
            ctxb, cur, ln1_g + i * HH, ln1_b + i * HH, cur);

        // FF
        if_gemm_wmma<<<dim3(FFF / 32, (BB * CC + 63) / 64), 256, 0, stream>>>(
            cur, W1 + (size_t)i * HH * FFF, b1 + i * FFF, f1b, BB * CC, FFF, HH, 1);
        if_gemm_wmma<<<dim3(HH / 32, (BB * CC + 63) / 64), 256, 0, stream>>>(
            f1b, W2 + (size_t)i * FFF * HH, b2 + i * HH, f2b, BB * CC, HH, FFF, 0);

        // out = LN2(ff + out); final layer writes d_out
        if_add_ln<<<BB * CC, 128, 0, stream>>>(
            f2b, cur, ln2_g + i * HH, ln2_b + i * HH, (i == LL - 1) ? out : cur);
    }
}